// DenseSigmoidFlow_47218870452708
// MI455X (gfx1250) — compile-verified
//
#include <hip/hip_runtime.h>
#include <hip/hip_bf16.h>
#include <math.h>

typedef __attribute__((ext_vector_type(2))) float v2f;
typedef __attribute__((ext_vector_type(8))) float v8f;

#define EPS_F    1e-6f
#define LOG100_F 4.6051701859880914f
#define LOG2E_F  1.4426950408889634f
#define LN2_F    0.6931471805599453f

// Raw-HW transcendentals: single v_exp_f32 / v_log_f32 / v_rcp_f32, no range guards.
// All call sites feed values in the safe (normal, finite) range.
__device__ __forceinline__ float fexp(float x) {
    return __builtin_amdgcn_exp2f(x * LOG2E_F);
}
__device__ __forceinline__ float flog(float x) {
    return __builtin_amdgcn_logf(x) * LN2_F;
}
__device__ __forceinline__ float frcp(float x) {
    return __builtin_amdgcn_rcpf(x);
}

__device__ __forceinline__ float fast_log1p(float t) {   // t in [0, 1]
    // small-t polynomial avoids catastrophic cancellation in flog(1+t)
    return (t < 0.03f) ? (t - 0.5f * t * t) : flog(1.f + t);
}
__device__ __forceinline__ float softplus_f(float z) {
    return fmaxf(z, 0.f) + fast_log1p(fexp(-fabsf(z)));
}
__device__ __forceinline__ float logsig_f(float z) {     // -softplus(-z) + EPS
    return -softplus_f(-z) + EPS_F;
}

// Cross-lane XOR butterfly via DS_SWIZZLE (group-of-32: offset = xor<<10 | and 0x1f)
template <int XMASK>
__device__ __forceinline__ float swz_xor_add(float x) {
    float o = __int_as_float(
        __builtin_amdgcn_ds_swizzle(__float_as_int(x), (XMASK << 10) | 0x1f));
    return x + o;
}

// One wave32 per (b,n); 8 waves (one b) per 256-thread block. Grid = B = 4096.
__global__ __launch_bounds__(256)
void dsf_kernel(const float* __restrict__ dsparams,  // (B, 8, 64)
                const float* __restrict__ xg,        // (B, 16)
                const float* __restrict__ logdet_in, // (B,)
                const float* __restrict__ u_,        // (16, 16)
                const float* __restrict__ w_,        // (16, 16)
                float* __restrict__ xnew,            // (B, 16)
                float* __restrict__ ldet)            // (B, 8, 16)
{
    __shared__ float dsb[512];          // dsparams for this b (8 * 64)
    __shared__ float xv[16];
    __shared__ float ubuf[256];
    __shared__ float wbuf[256];
    __shared__ float sUs [8][16 * 17];  // u-softmax, row-padded (bank-conflict-free)
    __shared__ float sWsm[8][16 * 17];  // w-softmax, row-padded
    __shared__ float sSig [8][16];
    __shared__ float sExpc[8][16];      // exp(c[h])
    __shared__ float sKc  [8][16];      // log(1-eps) - log(xpc) - log(1-xpc)

    const int t    = threadIdx.x;
    const int b    = blockIdx.x;
    const int warp = t >> 5;            // = n (0..7)
    const int lane = t & 31;
    const int l16  = lane & 15;
    const int half = lane >> 4;

    // Block-wide coalesced staging
    dsb[t]       = dsparams[b * 512 + t];
    dsb[t + 256] = dsparams[b * 512 + 256 + t];
    ubuf[t] = u_[t];
    wbuf[t] = w_[t];
    if (t < 16) xv[t] = xg[b * 16 + t];
    __syncthreads();

    const int n = warp;
    const float* ds  = &dsb[n * 64];
    float* Us  = sUs[warp];
    float* Wsm = sWsm[warp];

    // ---- Phase A: row softmaxes. lanes 0-15: U row h=l16; lanes 16-31: W row o=l16
    {
        const float* mrow = half ? &wbuf[l16 * 16] : &ubuf[l16 * 16];
        const float* add  = half ? &ds[32] : &ds[48];
        float e[16];
        float mx = -INFINITY;
        #pragma unroll
        for (int j = 0; j < 16; ++j) { float v = mrow[j] + add[j]; e[j] = v; mx = fmaxf(mx, v); }
        float s = 0.f;
        #pragma unroll
        for (int j = 0; j < 16; ++j) { float v = fexp(e[j] - mx); e[j] = v; s += v; }
        float inv_s = frcp(s);
        float* dst = half ? &Wsm[l16 * 17] : &Us[l16 * 17];
        #pragma unroll
        for (int j = 0; j < 16; ++j) dst[j] = e[j] * inv_s;
    }
    __syncthreads();

    // ---- Phase B: lanes 0-15 handle h=l16: pre_sigm, sigmoid, c[h]
    if (half == 0) {
        const float invc = 0.54132324f;   // log(exp(1-EPS) - 1)
        float aH = softplus_f(ds[l16] + invc) + EPS_F;
        float bH = ds[16 + l16];
        float s = 0.f;
        #pragma unroll
        for (int i = 0; i < 16; ++i) s += Us[l16 * 17 + i] * xv[i];
        float ps = aH * s + bH;
        sSig[warp][l16] = frcp(1.f + fexp(-ps));
        float c = logsig_f(ps) + logsig_f(-ps) + (flog(aH * 100.f) - LOG100_F);
        sExpc[warp][l16] = fexp(c);
    }
    __syncthreads();

    // ---- Phase C: lanes 0-15 handle o=l16: x_pre, xnew (n==0), per-o constant
    if (half == 0) {
        float xp = 0.f;
        #pragma unroll
        for (int h = 0; h < 16; ++h) xp += Wsm[l16 * 17 + h] * sSig[warp][h];
        float xpc = xp * (1.f - EPS_F) + EPS_F * 0.5f;
        float lx = flog(xpc * 100.f) - LOG100_F;
        float l1 = flog((1.f - xpc) * 100.f) - LOG100_F;
        sKc[warp][l16] = flog(1.f - EPS_F) - lx - l1;
        if (n == 0) xnew[b * 16 + l16] = lx - l1;
    }
    __syncthreads();

    // ---- Phase D: P[o,i] = sum_h Ej[o,h] * Us[h,i]  via 4x V_WMMA_F32_16X16X4_F32
    // A (16x4 f32): lane row M=l16, VGPR0/1 = K = 2*half, 2*half+1 (per K-chunk)
    // B (4x16 f32): lane col N=l16, VGPR0/1 = K = 2*half, 2*half+1
    v8f acc = {};
    #pragma unroll
    for (int k0 = 0; k0 < 16; k0 += 4) {
        int kk = k0 + 2 * half;
        v2f av, bv;
        av[0] = Wsm[l16 * 17 + kk]     * sExpc[warp][kk];
        av[1] = Wsm[l16 * 17 + kk + 1] * sExpc[warp][kk + 1];
        bv[0] = Us[kk * 17 + l16];
        bv[1] = Us[(kk + 1) * 17 + l16];
        acc = __builtin_amdgcn_wmma_f32_16x16x4_f32(
            false, av, false, bv, (short)0, acc, false, false);
    }

    // D layout: VGPR v -> row o = v + 8*half, lane -> column i.
    // Reduce over i = 16-lane XOR butterfly (DS_SWIZZLE SWAPX1/2/4/8).
    const float lda = logdet_in[b];
    float* ldo = &ldet[(b * 8 + n) * 16];
    #pragma unroll
    for (int v = 0; v < 8; ++v) {
        float r = acc[v];
        r = swz_xor_add<1>(r);
        r = swz_xor_add<2>(r);
        r = swz_xor_add<4>(r);
        r = swz_xor_add<8>(r);
        int o = v + 8 * half;
        if (l16 == 0) ldo[o] = flog(r) + sKc[warp][o] + lda;
    }
}

extern "C" void kernel_launch(void* const* d_in, const int* in_sizes, int n_in,
                              void* d_out, int out_size, void* d_ws, size_t ws_size,
                              hipStream_t stream) {
    const float* dsparams = (const float*)d_in[0];
    const float* x        = (const float*)d_in[1];
    const float* logdet   = (const float*)d_in[2];
    const float* u_       = (const float*)d_in[3];
    const float* w_       = (const float*)d_in[4];

    float* out   = (float*)d_out;
    float* xnew  = out;                 // (4096, 16)
    float* ldet  = out + 4096 * 16;     // (4096, 8, 16, 1)

    dsf_kernel<<<4096, 256, 0, stream>>>(dsparams, x, logdet, u_, w_, xnew, ldet);
}